// BeeAlgorithm_50964081934652
// MI455X (gfx1250) — compile-verified
//
#include <hip/hip_runtime.h>
#include <math.h>

typedef __attribute__((ext_vector_type(2))) float v2f;
typedef __attribute__((ext_vector_type(8))) float v8f;

#define HIDDEN 1024
#define NBEES  40
#define NEMP   20
#define NONL   20
#define NSCOUT 4
#define ROWS   32768   // B*S = 4*8192
#define NCOL   64      // 40 bee cols + 20 onlooker-noise cols + 4 junk pad

// ---- workspace layout (float offsets) ----
#define WS_D        0                        // ROWS*NCOL     (8 MB)
#define WS_X2       (ROWS*NCOL)              // ROWS
#define WS_DCOL     (WS_X2 + ROWS)           // ROWS
#define WS_POS      (WS_DCOL + ROWS)         // NBEES*HIDDEN
#define WS_BESTVEC  (WS_POS + NBEES*HIDDEN)  // HIDDEN (16B aligned)
#define WS_FIT      (WS_BESTVEC + HIDDEN)    // NBEES
#define WS_SEL      (WS_FIT + NBEES)         // NONL ints
#define WS_BESTFIT  (WS_SEL + NONL)          // 1

// ============================================================================
// K1: D[r, 0:40] = x_row . bee_pos[c];  D[r, 40:60] = x_row . onlooker_noise
//     plus x2[r] = ||x_row||^2.   WMMA f32 16x16x4, one wave per 16-row tile.
// ============================================================================
__global__ __launch_bounds__(256) void k1_gemm(const float* __restrict__ x,
                                               const float* __restrict__ bees,
                                               const float* __restrict__ onoise,
                                               float* __restrict__ D,
                                               float* __restrict__ x2)
{
  const int lane = threadIdx.x & 31;
  const int wave = threadIdx.x >> 5;
  const int tile = blockIdx.x * 8 + wave;          // 2048 tiles of 16 rows
  const int m    = lane & 15;
  const int row  = tile * 16 + m;
  const int kk   = (lane >> 4) * 2;                // lanes 0-15: K={0,1}; 16-31: K={2,3}

  // per-N-tile B column pointers (cols >=60 clamped; their D cols are junk, never read)
  const float* bp[4];
  #pragma unroll
  for (int t = 0; t < 4; ++t) {
    int col = t * 16 + m;
    int c = col < 59 ? col : 59;
    bp[t] = (c < NBEES) ? (bees + c * HIDDEN) : (onoise + (c - NBEES) * HIDDEN);
  }
  const float* ap = x + (size_t)row * HIDDEN + kk;

  v8f acc[4] = {};
  float ssq = 0.f;
  for (int k = 0; k < HIDDEN; k += 4) {
    v2f a = *(const v2f*)(ap + k);                 // A 16x4 f32 fragment (2 VGPRs)
    ssq += a.x * a.x + a.y * a.y;
    #pragma unroll
    for (int t = 0; t < 4; ++t) {
      v2f b = *(const v2f*)(bp[t] + k + kk);       // B 4x16 f32 fragment (2 VGPRs)
      acc[t] = __builtin_amdgcn_wmma_f32_16x16x4_f32(false, a, false, b,
                                                     (short)0, acc[t], false, false);
    }
  }
  // lanes L and L+16 together cover the full row -> combine for x2
  ssq += __shfl_xor(ssq, 16);
  if (lane < 16) x2[row] = ssq;

  // C/D layout: VGPR r -> M=r (lanes 0-15) / M=r+8 (lanes 16-31), N=lane&15
  const int rbase = tile * 16 + ((lane >> 4) * 8);
  #pragma unroll
  for (int t = 0; t < 4; ++t) {
    const int col = t * 16 + m;
    #pragma unroll
    for (int r = 0; r < 8; ++r)
      D[(size_t)(rbase + r) * NCOL + col] = acc[t][r];
  }
}

// ============================================================================
// K2: phase-1 fitness of all 40 bees from D + x2 (one block per bee)
// ============================================================================
__global__ __launch_bounds__(256) void k2_fit(const float* __restrict__ bees,
                                              const float* __restrict__ D,
                                              const float* __restrict__ x2,
                                              float* __restrict__ fitv)
{
  __shared__ float red[256];
  const int n = blockIdx.x, tid = threadIdx.x;
  float p2 = 0.f;
  for (int h = tid; h < HIDDEN; h += 256) { float v = bees[n * HIDDEN + h]; p2 += v * v; }
  red[tid] = p2; __syncthreads();
  for (int off = 128; off > 0; off >>= 1) { if (tid < off) red[tid] += red[tid + off]; __syncthreads(); }
  p2 = red[0]; __syncthreads();
  float s = 0.f;
  for (int r = tid; r < ROWS; r += 256) {
    float v = x2[r] - 2.f * D[(size_t)r * NCOL + n] + p2;
    s += sqrtf(fmaxf(v, 0.f));
  }
  red[tid] = s; __syncthreads();
  for (int off = 128; off > 0; off >>= 1) { if (tid < off) red[tid] += red[tid + off]; __syncthreads(); }
  if (tid == 0) fitv[n] = red[0] * (1.f / ROWS);
}

// ============================================================================
// K3: single-block sequential core: best-bee update, 20 employed steps
//     (linear dot updates on D columns), Gumbel-max onlooker selection.
//     Global RAW hazards are all same-thread (r ≡ tid mod 1024 everywhere).
// ============================================================================
__global__ __launch_bounds__(1024) void k3_seq(const float* __restrict__ bees,
    const float* __restrict__ bestp_in, const float* __restrict__ bestf_in,
    const float* __restrict__ er_p, const int* __restrict__ nbr,
    const float* __restrict__ gumbel,
    float* __restrict__ D, const float* __restrict__ x2,
    float* __restrict__ dcol, float* __restrict__ pos_ws,
    float* __restrict__ bestvec, float* __restrict__ fitv,
    int* __restrict__ sel, float* __restrict__ bestfit,
    float* __restrict__ out_bp, float* __restrict__ out_bf)
{
  __shared__ float red[1024];
  __shared__ float fs[NBEES];
  __shared__ int s_best, s_imp, s_take;
  __shared__ float s_c2;
  const int tid = threadIdx.x;

  if (tid < NBEES) fs[tid] = fitv[tid];
  __syncthreads();
  // ---- phase 1 decision ----
  if (tid == 0) {
    int bi = 0; float bv = fs[0];
    for (int j = 1; j < NBEES; ++j) if (fs[j] > bv) { bv = fs[j]; bi = j; }
    float bf0 = bestf_in[0];
    int imp = bv > bf0 ? 1 : 0;
    s_best = bi; s_imp = imp;
    float nb = imp ? bv : bf0;
    bestfit[0] = nb; out_bf[0] = nb;
  }
  __syncthreads();
  {
    float v = s_imp ? bees[s_best * HIDDEN + tid] : bestp_in[tid];
    bestvec[tid] = v;
    out_bp[tid] = v;
  }
  for (int j = tid; j < NBEES * HIDDEN; j += 1024) pos_ws[j] = bees[j];
  __syncthreads();

  // ---- phase 2: employed bees (sequential) ----
  const float er = er_p[0];
  const float aa = 1.f + er;
  for (int i = 0; i < NEMP; ++i) {
    const int nb = nbr[i];
    float cand = aa * pos_ws[i * HIDDEN + tid] - er * pos_ws[nb * HIDDEN + tid];
    red[tid] = cand * cand;
    __syncthreads();
    for (int off = 512; off > 0; off >>= 1) { if (tid < off) red[tid] += red[tid + off]; __syncthreads(); }
    if (tid == 0) s_c2 = red[0];
    __syncthreads();
    const float c2 = s_c2;
    float s = 0.f;
    for (int r = tid; r < ROWS; r += 1024) {
      float dc = aa * D[(size_t)r * NCOL + i] - er * D[(size_t)r * NCOL + nb];
      dcol[r] = dc;
      float v = x2[r] - 2.f * dc + c2;
      s += sqrtf(fmaxf(v, 0.f));
    }
    red[tid] = s;
    __syncthreads();
    for (int off = 512; off > 0; off >>= 1) { if (tid < off) red[tid] += red[tid + off]; __syncthreads(); }
    if (tid == 0) {
      float nf = red[0] * (1.f / ROWS);
      s_take = nf > fs[i] ? 1 : 0;
      if (s_take) fs[i] = nf;
    }
    __syncthreads();
    if (s_take) {
      pos_ws[i * HIDDEN + tid] = cand;
      for (int r = tid; r < ROWS; r += 1024) D[(size_t)r * NCOL + i] = dcol[r];
    }
    __syncthreads();
  }

  if (tid < NBEES) fitv[tid] = fs[tid];
  // Gumbel-max: argmax(log_softmax(f)+g) == argmax(f+g) (constant shift drops)
  if (tid < NONL) {
    int bj = 0; float bv = fs[0] + gumbel[tid * NEMP + 0];
    for (int j = 1; j < NEMP; ++j) {
      float v = fs[j] + gumbel[tid * NEMP + j];
      if (v > bv) { bv = v; bj = j; }
    }
    sel[tid] = bj;
  }
}

// ============================================================================
// K4: onlookers (independent, one block each)
// ============================================================================
__global__ __launch_bounds__(256) void k4_onl(const float* __restrict__ onoise,
    const float* __restrict__ xr_p, const float* __restrict__ D,
    const float* __restrict__ x2, float* __restrict__ pos_ws,
    float* __restrict__ fitv, const int* __restrict__ sel)
{
  __shared__ float red[256];
  __shared__ float s_c2;
  __shared__ int s_take;
  const int o = blockIdx.x, tid = threadIdx.x;
  const int so = sel[o];
  const float xr = xr_p[0];
  float cand[4]; float ssq = 0.f;
  #pragma unroll
  for (int u = 0; u < 4; ++u) {
    int h = tid + u * 256;
    float c = pos_ws[so * HIDDEN + h] + xr * onoise[o * HIDDEN + h];
    cand[u] = c; ssq += c * c;
  }
  red[tid] = ssq; __syncthreads();
  for (int off = 128; off > 0; off >>= 1) { if (tid < off) red[tid] += red[tid + off]; __syncthreads(); }
  if (tid == 0) s_c2 = red[0];
  __syncthreads();
  const float c2 = s_c2;
  float s = 0.f;
  for (int r = tid; r < ROWS; r += 256) {
    float dc = D[(size_t)r * NCOL + so] + xr * D[(size_t)r * NCOL + NBEES + o];
    float v = x2[r] - 2.f * dc + c2;
    s += sqrtf(fmaxf(v, 0.f));
  }
  red[tid] = s; __syncthreads();
  for (int off = 128; off > 0; off >>= 1) { if (tid < off) red[tid] += red[tid + off]; __syncthreads(); }
  if (tid == 0) {
    float nf = red[0] * (1.f / ROWS);
    s_take = nf > fitv[NEMP + o] ? 1 : 0;
    if (s_take) fitv[NEMP + o] = nf;
  }
  __syncthreads();
  if (s_take) {
    #pragma unroll
    for (int u = 0; u < 4; ++u) pos_ws[(NEMP + o) * HIDDEN + tid + u * 256] = cand[u];
  }
}

// ============================================================================
// K5: scouts (4 worst, stable ascending) + write positions/fitness outputs
// ============================================================================
__global__ __launch_bounds__(1024) void k5_final(const float* __restrict__ scout,
    const float* __restrict__ pos_ws, const float* __restrict__ fitv,
    float* __restrict__ out_pos, float* __restrict__ out_fit)
{
  __shared__ float fs[NBEES];
  __shared__ int rep[NBEES];
  const int tid = threadIdx.x;
  if (tid < NBEES) { fs[tid] = fitv[tid]; rep[tid] = 0; }
  __syncthreads();
  if (tid == 0) {
    int used[NBEES] = {0};
    for (int s = 0; s < NSCOUT; ++s) {
      int bi = -1; float bv = 0.f;
      for (int j = 0; j < NBEES; ++j) {
        if (used[j]) continue;
        if (bi < 0 || fs[j] < bv) { bv = fs[j]; bi = j; }
      }
      used[bi] = 1; rep[bi] = s + 1;
    }
    for (int j = 0; j < NBEES; ++j) if (rep[j]) fs[j] = 0.f;
  }
  __syncthreads();
  if (tid < NBEES) out_fit[tid] = fs[tid];
  for (int j = tid; j < NBEES * HIDDEN; j += 1024) {
    int bee = j >> 10, h = j & 1023;
    int r = rep[bee];
    out_pos[j] = r ? scout[(r - 1) * HIDDEN + h] * 0.1f : pos_ws[j];
  }
}

// ============================================================================
// K6: output = x + best_position (float4 streaming, 268 MB of HBM traffic)
// ============================================================================
__global__ __launch_bounds__(256) void k6_out(const float4* __restrict__ x4,
                                              const float* __restrict__ bestvec,
                                              float4* __restrict__ o4)
{
  const int i = blockIdx.x * 256 + threadIdx.x;     // float4 index
  const float4* b4 = (const float4*)bestvec;
  float4 xv = x4[i];
  float4 bv = b4[i & 255];                          // 1024 floats = 256 float4 per row
  xv.x += bv.x; xv.y += bv.y; xv.z += bv.z; xv.w += bv.w;
  o4[i] = xv;
}

extern "C" void kernel_launch(void* const* d_in, const int* in_sizes, int n_in,
                              void* d_out, int out_size, void* d_ws, size_t ws_size,
                              hipStream_t stream) {
  (void)in_sizes; (void)n_in; (void)out_size; (void)ws_size;
  const float* x        = (const float*)d_in[0];
  const float* bees     = (const float*)d_in[1];
  /* d_in[2] bee_fitness unused: reference overwrites it in phase 1 */
  const float* bestp_in = (const float*)d_in[3];
  const float* bestf_in = (const float*)d_in[4];
  const float* er_p     = (const float*)d_in[5];
  const float* xr_p     = (const float*)d_in[6];
  const int*   nbr      = (const int*)d_in[7];
  const float* gumbel   = (const float*)d_in[8];
  const float* onoise   = (const float*)d_in[9];
  const float* scout    = (const float*)d_in[10];

  float* ws      = (float*)d_ws;
  float* D       = ws + WS_D;
  float* x2      = ws + WS_X2;
  float* dcol    = ws + WS_DCOL;
  float* pos_ws  = ws + WS_POS;
  float* bestvec = ws + WS_BESTVEC;
  float* fitv    = ws + WS_FIT;
  int*   sel     = (int*)(ws + WS_SEL);
  float* bestfit = ws + WS_BESTFIT;

  float* out     = (float*)d_out;
  float* out_pos = out + (size_t)ROWS * HIDDEN;   // 33554432
  float* out_fit = out_pos + NBEES * HIDDEN;      // +40960
  float* out_bp  = out_fit + NBEES;               // +40
  float* out_bf  = out_bp + HIDDEN;               // +1024

  k1_gemm<<<256, 256, 0, stream>>>(x, bees, onoise, D, x2);
  k2_fit<<<NBEES, 256, 0, stream>>>(bees, D, x2, fitv);
  k3_seq<<<1, 1024, 0, stream>>>(bees, bestp_in, bestf_in, er_p, nbr, gumbel,
                                 D, x2, dcol, pos_ws, bestvec, fitv, sel, bestfit,
                                 out_bp, out_bf);
  k4_onl<<<NONL, 256, 0, stream>>>(onoise, xr_p, D, x2, pos_ws, fitv, sel);
  k5_final<<<1, 1024, 0, stream>>>(scout, pos_ws, fitv, out_pos, out_fit);
  k6_out<<<(ROWS * HIDDEN) / 1024, 256, 0, stream>>>((const float4*)x, bestvec, (float4*)d_out);
}